// LocalSpatialEncoding_44212393345653
// MI455X (gfx1250) — compile-verified
//
#include <hip/hip_runtime.h>
#include <hip/hip_bf16.h>

typedef float v2f __attribute__((ext_vector_type(2)));
typedef float v8f __attribute__((ext_vector_type(8)));

static constexpr int B = 4, N = 65536, K = 16, D = 16;
static constexpr int NPOINTS = B * N;                  // 262144 points
static constexpr size_t PLANE = (size_t)N * K;         // elements per (b,channel) plane
static constexpr float EPS = 1e-6f;

// workspace float layout:
// [0..15]  per-channel sum
// [16..31] per-channel sum of squares
// [32..47] BN scale
// [48..63] BN shift
// [64 ..]  transposed features (B,N,16) = 4M floats
static constexpr int WS_SUM = 0, WS_SQ = 16, WS_SCALE = 32, WS_SHIFT = 48, WS_FEATT = 64;

__global__ void k_init(float* __restrict__ ws) {
    if (threadIdx.x < 32) ws[threadIdx.x] = 0.0f;
}

// One wave = one point n per iteration.
// A = conv_w (16x12, zero padded), B = G (12x16 geometric features), C = bias.
// D tile: lane l, vgpr v -> channel o = v + 8*(l>=16), neighbor k = l%16.
__global__ __launch_bounds__(256) void k_conv(
    const float* __restrict__ coords, const int* __restrict__ nbr,
    const float* __restrict__ conv_w, const float* __restrict__ conv_b,
    float* __restrict__ out, float* __restrict__ ws, int wavesTotal) {
    __shared__ float lsum[16];
    __shared__ float lsq[16];
    const int tid = threadIdx.x;
    if (tid < 16) lsum[tid] = 0.0f;
    else if (tid < 32) lsq[tid - 16] = 0.0f;
    __syncthreads();

    const int lane = tid & 31;
    const int k = lane & 15;     // A-row (channel M) AND B-column (neighbor N)
    const int hi = lane >> 4;
    const int wave = blockIdx.x * (blockDim.x >> 5) + (tid >> 5);

    // A matrix chunks: chunk c, vgpr v, lane -> K = 4c + v + 2*hi ; row M = k
    v2f A0, A1, A2;
    A0.x = conv_w[k * 10 + 0 + 2 * hi];
    A0.y = conv_w[k * 10 + 1 + 2 * hi];
    A1.x = conv_w[k * 10 + 4 + 2 * hi];
    A1.y = conv_w[k * 10 + 5 + 2 * hi];
    A2.x = hi ? 0.0f : conv_w[k * 10 + 8];
    A2.y = hi ? 0.0f : conv_w[k * 10 + 9];

    float bias[8];
#pragma unroll
    for (int v = 0; v < 8; ++v) bias[v] = conv_b[v + 8 * hi];

    float accS[8], accQ[8];
#pragma unroll
    for (int v = 0; v < 8; ++v) { accS[v] = 0.0f; accQ[v] = 0.0f; }

    for (int p = wave; p < NPOINTS; p += wavesTotal) {   // uniform per wave: EXEC stays full
        const int b = p >> 16;
        const int n = p & (N - 1);
        const float* pc = coords + (size_t)p * 3;
        const float px = pc[0], py = pc[1], pz = pc[2];
        const int idxv = nbr[(size_t)p * 16 + k];
        const float* qc = coords + ((size_t)b * N + idxv) * 3;
        const float qx = qc[0], qy = qc[1], qz = qc[2];
        const float rx = px - qx, ry = py - qy, rz = pz - qz;
        const float dist = __builtin_amdgcn_sqrtf(rx * rx + ry * ry + rz * rz);

        // G rows: 0=dist 1=rx 2=ry 3=rz 4=px 5=py 6=pz 7=qx 8=qy 9=qz, 10..11 = 0
        v2f B0, B1, B2;
        B0.x = hi ? ry : dist;  B0.y = hi ? rz : rx;     // rows {0,1} / {2,3}
        B1.x = hi ? pz : px;    B1.y = hi ? qx : py;     // rows {4,5} / {6,7}
        B2.x = hi ? 0.0f : qy;  B2.y = hi ? 0.0f : qz;   // rows {8,9} / {10,11}

        v8f c;
#pragma unroll
        for (int v = 0; v < 8; ++v) c[v] = bias[v];
        c = __builtin_amdgcn_wmma_f32_16x16x4_f32(false, A0, false, B0, (short)0, c, false, false);
        c = __builtin_amdgcn_wmma_f32_16x16x4_f32(false, A1, false, B1, (short)0, c, false, false);
        c = __builtin_amdgcn_wmma_f32_16x16x4_f32(false, A2, false, B2, (short)0, c, false, false);

        // lane base for channel (16 + 8*hi + v): constant 4MB stride per v
        float* xo = out + (((size_t)(b * 32 + 16 + 8 * hi) * N + n) * 16 + k);
#pragma unroll
        for (int v = 0; v < 8; ++v) {
            const float val = c[v];
            xo[(size_t)v * PLANE] = val;          // pre-BN x in final slot
            accS[v] += val;
            accQ[v] += val * val;
        }
    }

    // butterfly reduce across the 16 lanes of each half (channels differ only by hi)
#pragma unroll
    for (int m = 8; m >= 1; m >>= 1) {
#pragma unroll
        for (int v = 0; v < 8; ++v) {
            accS[v] += __shfl_xor(accS[v], m, 16);
            accQ[v] += __shfl_xor(accQ[v], m, 16);
        }
    }
    if (k == 0) {
#pragma unroll
        for (int v = 0; v < 8; ++v) {
            atomicAdd(&lsum[v + 8 * hi], accS[v]);
            atomicAdd(&lsq[v + 8 * hi], accQ[v]);
        }
    }
    __syncthreads();
    if (tid < 16) atomicAdd(&ws[WS_SUM + tid], lsum[tid]);
    else if (tid < 32) atomicAdd(&ws[WS_SQ + tid - 16], lsq[tid - 16]);
}

__global__ void k_stats(const float* __restrict__ gamma, const float* __restrict__ beta,
                        float* __restrict__ ws) {
    const int o = threadIdx.x;
    if (o < 16) {
        const float cnt = (float)((size_t)B * N * K);
        const float mean = ws[WS_SUM + o] / cnt;
        const float var = ws[WS_SQ + o] / cnt - mean * mean;
        const float sc = gamma[o] * rsqrtf(var + EPS);
        ws[WS_SCALE + o] = sc;
        ws[WS_SHIFT + o] = beta[o] - mean * sc;
    }
}

// features (B,D,N) -> featT (B,N,D) so neighbor gathers read one 64B row
__global__ __launch_bounds__(256) void k_transpose(const float* __restrict__ feat,
                                                   float* __restrict__ featT) {
    const size_t t = (size_t)blockIdx.x * blockDim.x + threadIdx.x;  // over B*N*D
    const int d = (int)(t & 15);
    const size_t pn = t >> 4;
    const int b = (int)(pn >> 16);
    const int n = (int)(pn & 65535);
    featT[t] = feat[((size_t)b * 16 + d) * N + n];
}

// fused BN+ReLU in place + neighbor-feature gather into channels 0..15
__global__ __launch_bounds__(256) void k_final(
    const float* __restrict__ featT, const int* __restrict__ nbr,
    const float* __restrict__ ws, float* __restrict__ out) {
    __shared__ float sc[16], sh[16];
    if (threadIdx.x < 16) sc[threadIdx.x] = ws[WS_SCALE + threadIdx.x];
    else if (threadIdx.x < 32) sh[threadIdx.x - 16] = ws[WS_SHIFT + threadIdx.x - 16];
    __syncthreads();

    const size_t t = (size_t)blockIdx.x * blockDim.x + threadIdx.x;  // over B*N*K
    const int k = (int)(t & 15);
    const size_t pn = t >> 4;
    const int b = (int)(pn >> 16);
    const int n = (int)(pn & 65535);
    const int idxv = nbr[t];

    const float4* fr = (const float4*)(featT + ((size_t)b * N + idxv) * 16);
    float f[16];
    float4 f4;
#pragma unroll
    for (int q = 0; q < 4; ++q) {
        f4 = fr[q];
        f[4 * q + 0] = f4.x; f[4 * q + 1] = f4.y; f[4 * q + 2] = f4.z; f[4 * q + 3] = f4.w;
    }
    float* go = out + (((size_t)(b * 32) * N + n) * 16 + k);       // channel 0 plane
    float* xo = go + 16 * PLANE;                                    // channel 16 plane
#pragma unroll
    for (int o = 0; o < 16; ++o) {
        float v = xo[(size_t)o * PLANE];
        v = fmaxf(fmaf(v, sc[o], sh[o]), 0.0f);
        xo[(size_t)o * PLANE] = v;
        go[(size_t)o * PLANE] = f[o];
    }
}

// fallback if workspace cannot hold the transposed features
__global__ __launch_bounds__(256) void k_final_direct(
    const float* __restrict__ feat, const int* __restrict__ nbr,
    const float* __restrict__ ws, float* __restrict__ out) {
    __shared__ float sc[16], sh[16];
    if (threadIdx.x < 16) sc[threadIdx.x] = ws[WS_SCALE + threadIdx.x];
    else if (threadIdx.x < 32) sh[threadIdx.x - 16] = ws[WS_SHIFT + threadIdx.x - 16];
    __syncthreads();

    const size_t t = (size_t)blockIdx.x * blockDim.x + threadIdx.x;
    const int k = (int)(t & 15);
    const size_t pn = t >> 4;
    const int b = (int)(pn >> 16);
    const int n = (int)(pn & 65535);
    const int idxv = nbr[t];
    float* go = out + (((size_t)(b * 32) * N + n) * 16 + k);
    float* xo = go + 16 * PLANE;
#pragma unroll
    for (int o = 0; o < 16; ++o) {
        float v = xo[(size_t)o * PLANE];
        v = fmaxf(fmaf(v, sc[o], sh[o]), 0.0f);
        xo[(size_t)o * PLANE] = v;
        go[(size_t)o * PLANE] = feat[((size_t)b * 16 + o) * N + idxv];
    }
}

extern "C" void kernel_launch(void* const* d_in, const int* in_sizes, int n_in,
                              void* d_out, int out_size, void* d_ws, size_t ws_size,
                              hipStream_t stream) {
    (void)in_sizes; (void)n_in; (void)out_size;
    const float* coords   = (const float*)d_in[0];
    const float* features = (const float*)d_in[1];
    const int*   nbr      = (const int*)d_in[2];
    const float* conv_w   = (const float*)d_in[3];
    const float* conv_b   = (const float*)d_in[4];
    const float* gamma    = (const float*)d_in[5];
    const float* beta     = (const float*)d_in[6];
    float* out = (float*)d_out;
    float* ws  = (float*)d_ws;

    k_init<<<1, 32, 0, stream>>>(ws);

    const int convBlocks = 2048;                       // 8 waves/block -> 16384 waves
    const int wavesTotal = convBlocks * (256 / 32);    // 16 points per wave
    k_conv<<<convBlocks, 256, 0, stream>>>(coords, nbr, conv_w, conv_b, out, ws, wavesTotal);

    k_stats<<<1, 32, 0, stream>>>(gamma, beta, ws);

    const size_t needWs = (size_t)(WS_FEATT + (size_t)B * N * D) * sizeof(float);
    const int elemBlocks = (B * N * K) / 256;          // 16384
    if (ws_size >= needWs) {
        k_transpose<<<(B * N * D) / 256, 256, 0, stream>>>(features, ws + WS_FEATT);
        k_final<<<elemBlocks, 256, 0, stream>>>(ws + WS_FEATT, nbr, ws, out);
    } else {
        k_final_direct<<<elemBlocks, 256, 0, stream>>>(features, nbr, ws, out);
    }
}